// input_masked_conv2d_4690104287876
// MI455X (gfx1250) — compile-verified
//
#include <hip/hip_runtime.h>

typedef __attribute__((ext_vector_type(2))) float v2f;
typedef __attribute__((ext_vector_type(8))) float v8f;

namespace {
constexpr int BATCH = 16;
constexpr int CIN   = 64;
constexpr int COUT  = 64;
constexpr int HIMG  = 64;
constexpr int WIMG  = 64;
constexpr int TAPS  = 9;                 // 3x3 kernel
constexpr int KDIM  = CIN * TAPS;        // 576 (GEMM K)
constexpr int PSTR  = 68;                // P row stride in floats (bank-skewed)
}

// One workgroup per (batch, image row). 4 wave32 per block; 2 blocks co-resident
// per WGP (2 x 157KB LDS < 320KB) so staging of one overlaps compute of the other.
// Phase 1: build masked im2col panel P[576][64] in LDS (mask pre-folded).
// Phase 2: each wave owns a 2x2 grid of 16x16 C tiles -> per K-step:
//          4 v_wmma_f32_16x16x4_f32 fed by 2 global_load_b64 + 2 ds_load_2addr.
__global__ __launch_bounds__(128)
void masked_conv2d_wmma_f32(const float* __restrict__ x,
                            const float* __restrict__ mask,
                            const float* __restrict__ wgt,
                            const float* __restrict__ bias,
                            float* __restrict__ out)
{
  __shared__ float P[KDIM * PSTR];       // 576*68*4 = 156,672 B

  const int tid = threadIdx.x;
  const int y   = blockIdx.x & (HIMG - 1);   // image row
  const int b   = blockIdx.x >> 6;           // batch

  const float* xb = x + (size_t)b * CIN * HIMG * WIMG;

  // ---- phase 1: P[k = c*9 + t][n] = x[c, y+i-1, n+j-1] * mask[t, y*64+n] ----
  // t fully unrolled: i, j, row predicate, mask value invariant per thread.
  const int n     = tid & (WIMG - 1);        // pixel col owned by this thread
  const int cbase = tid >> 6;                // 0..1
#pragma unroll
  for (int t = 0; t < TAPS; ++t) {
    const int  i    = t / 3;
    const int  j    = t - 3 * i;
    const int  row  = y + i - 1;             // wave-uniform
    const int  col  = n + j - 1;
    const bool ok   = ((unsigned)row < (unsigned)HIMG) &
                      ((unsigned)col < (unsigned)WIMG);
    const float mv  = mask[t * (HIMG * WIMG) + y * WIMG + n];
    const float* xsrc = xb + (size_t)row * WIMG + col;   // + c*4096 per channel
#pragma unroll 4
    for (int c = cbase; c < CIN; c += 2) {
      float v = 0.0f;
      if (ok) v = xsrc[(size_t)c * (HIMG * WIMG)] * mv;
      P[(c * TAPS + t) * PSTR + n] = v;
    }
  }
  __syncthreads();

  // ---- phase 2: out_tile[64 x 64] = W[64 x 576] * P[576 x 64] ----
  const int wave  = tid >> 5;           // 0..3
  const int lane  = tid & 31;
  const int half  = lane >> 4;          // 0: lanes 0-15, 1: lanes 16-31
  const int lidx  = lane & 15;
  const int mbase = (wave & 1) * 32;    // two M tiles: mbase, mbase+16
  const int nbase = (wave >> 1) * 32;   // two N tiles: nbase, nbase+16
  const int pix0  = nbase + lidx;
  const int pix1  = pix0 + 16;

  // A (16x4 f32): M = lane&15; VGPR v holds K = k0 + 2*half + v
  const int koff = half * 2;
  const float* wrow0 = wgt + (size_t)(mbase + lidx) * KDIM + koff;
  const float* wrow1 = wrow0 + (size_t)16 * KDIM;
  // B (4x16 f32): N = lane&15; VGPR v holds K = k0 + 2*half + v
  const float* pb = &P[koff * PSTR + pix0];

  v8f acc00 = {};   // (M tile 0, N tile 0)
  v8f acc01 = {};   // (M tile 0, N tile 1)
  v8f acc10 = {};   // (M tile 1, N tile 0)
  v8f acc11 = {};   // (M tile 1, N tile 1)

#pragma unroll 4
  for (int k0 = 0; k0 < KDIM; k0 += 4) {
    if ((k0 & 63) == 0) {
      __builtin_prefetch(wrow0 + k0 + 128, 0, 3);  // WGP-scope prefetch of A
      __builtin_prefetch(wrow1 + k0 + 128, 0, 3);
    }

    v2f a0, a1;
    a0.x = wrow0[k0];
    a0.y = wrow0[k0 + 1];
    a1.x = wrow1[k0];
    a1.y = wrow1[k0 + 1];

    v2f b0, b1;
    b0.x = pb[0];
    b0.y = pb[PSTR];          // ds_load_2addr_b32 candidates
    b1.x = pb[16];
    b1.y = pb[PSTR + 16];

    acc00 = __builtin_amdgcn_wmma_f32_16x16x4_f32(false, a0, false, b0,
                                                  (short)0, acc00, false, false);
    acc01 = __builtin_amdgcn_wmma_f32_16x16x4_f32(false, a0, false, b1,
                                                  (short)0, acc01, false, false);
    acc10 = __builtin_amdgcn_wmma_f32_16x16x4_f32(false, a1, false, b0,
                                                  (short)0, acc10, false, false);
    acc11 = __builtin_amdgcn_wmma_f32_16x16x4_f32(false, a1, false, b1,
                                                  (short)0, acc11, false, false);

    pb += 4 * PSTR;           // only pointer math in the loop
  }

  // ---- epilogue: add bias, store C tiles (VGPR r -> M = r + 8*half) ----
  float* ob = out + (size_t)b * COUT * HIMG * WIMG + (size_t)y * WIMG;
#pragma unroll
  for (int r = 0; r < 8; ++r) {
    int o0 = mbase + half * 8 + r;        // M tile 0
    int o1 = o0 + 16;                     // M tile 1
    float bv0 = bias[o0];
    float bv1 = bias[o1];
    ob[(size_t)o0 * (HIMG * WIMG) + pix0] = acc00[r] + bv0;
    ob[(size_t)o0 * (HIMG * WIMG) + pix1] = acc01[r] + bv0;
    ob[(size_t)o1 * (HIMG * WIMG) + pix0] = acc10[r] + bv1;
    ob[(size_t)o1 * (HIMG * WIMG) + pix1] = acc11[r] + bv1;
  }
}

extern "C" void kernel_launch(void* const* d_in, const int* in_sizes, int n_in,
                              void* d_out, int out_size, void* d_ws, size_t ws_size,
                              hipStream_t stream) {
  (void)in_sizes; (void)n_in; (void)out_size; (void)d_ws; (void)ws_size;
  const float* x    = (const float*)d_in[0];   // [16,64,64,64]
  const float* mask = (const float*)d_in[1];   // [9,4096]
  const float* wgt  = (const float*)d_in[2];   // [64,64,3,3] == [64,576]
  const float* bias = (const float*)d_in[3];   // [64]
  float* out = (float*)d_out;                  // [16,64,64,64]

  dim3 grid(BATCH * HIMG);   // 1024 blocks: (batch, image row)
  dim3 block(128);           // 4 wave32; 2 blocks per WGP
  masked_conv2d_wmma_f32<<<grid, block, 0, stream>>>(x, mask, wgt, bias, out);
}